// EvolveGNN_O_53266184405474
// MI455X (gfx1250) — compile-verified
//
#include <hip/hip_runtime.h>
#include <hip/hip_bf16.h>
#include <math.h>

// ---------------------------------------------------------------------------
// EvolveGNN-O for MI455X (gfx1250, wave32, WMMA).
//   out = Ahat @ (x @ W^T) + bias,  W generated by a 1-step GRU.
// Dominant cost: 1.6M-edge gather/scatter (~0.85 GB irregular, L2-resident).
// Dense GEMM (100k x 64 x 64, fp32) mapped to V_WMMA_F32_16X16X4_F32.
// This revision: pair-interleaved LDS layouts so every WMMA fragment is one
// aligned ds_load_b64 (no operand-assembly v_movs), and 2 A-tiles per wave
// (M=32) so each B fragment feeds two WMMAs (8 back-to-back WMMAs / K-step).
// ---------------------------------------------------------------------------

#define N_NODES 100000
#define N_EDGES 1600000
#define CH      64            // IN_CH == OUT_CH == MEM == 64

#define WP_STRIDE 160         // B kpair-row stride in floats; 160 % 64 == 32
                              // -> half-waves land on disjoint bank halves

typedef __attribute__((ext_vector_type(2))) float v2f;
typedef __attribute__((ext_vector_type(8))) float v8f;

__device__ __forceinline__ float sigmoidf_(float x) { return 1.0f / (1.0f + __expf(-x)); }

// ---------------------------------------------------------------------------
// Kernel 1: GRU single step (h0 = 0) + generated weight  W[o*64+i]  (64x64).
// One block, 256 threads. Total work ~270K MACs -> microseconds.
// ---------------------------------------------------------------------------
__global__ void k_gen_weights(const float* __restrict__ mem,      // [64]
                              const float* __restrict__ w_ih,     // [192,64]
                              const float* __restrict__ b_ih,     // [192]
                              const float* __restrict__ b_hh,     // [192]
                              const float* __restrict__ wt_w,     // [4096,64]
                              const float* __restrict__ wt_b,     // [4096]
                              float* __restrict__ W)              // [4096] out
{
    __shared__ float s_m[CH];
    __shared__ float s_gi[3 * CH];
    __shared__ float s_um[CH];
    const int t = threadIdx.x;

    if (t < CH) s_m[t] = mem[t];
    __syncthreads();

    if (t < 3 * CH) {
        float acc = b_ih[t];
        const float* wr = w_ih + t * CH;
        #pragma unroll 8
        for (int k = 0; k < CH; ++k) acc += wr[k] * s_m[k];
        s_gi[t] = acc;
    }
    __syncthreads();

    if (t < CH) {
        float r  = sigmoidf_(s_gi[t]          + b_hh[t]);
        float z  = sigmoidf_(s_gi[CH + t]     + b_hh[CH + t]);
        float nn = tanhf    (s_gi[2 * CH + t] + r * b_hh[2 * CH + t]);
        s_um[t] = (1.0f - z) * nn;            // h' = (1-z)*n + z*0
    }
    __syncthreads();

    for (int i = t; i < CH * CH; i += blockDim.x) {
        float acc = wt_b[i];
        const float* wr = wt_w + i * CH;
        #pragma unroll 8
        for (int k = 0; k < CH; ++k) acc += wr[k] * s_um[k];
        W[i] = acc;                           // W[o][i], row-major [out, in]
    }
}

// ---------------------------------------------------------------------------
// Kernel 2: zero d_out (N*64) and deg (N). Must run every call (poisoning).
// ---------------------------------------------------------------------------
__global__ void k_zero(float* __restrict__ out, float* __restrict__ deg)
{
    int i = blockIdx.x * blockDim.x + threadIdx.x;
    if (i < N_NODES * CH) out[i] = 0.0f;
    if (i < N_NODES)      deg[i] = 0.0f;
}

// ---------------------------------------------------------------------------
// Kernel 3: degree of destination nodes (float atomics; exact for < 2^24).
// ---------------------------------------------------------------------------
__global__ void k_degree(const int* __restrict__ ei, float* __restrict__ deg)
{
    int e = blockIdx.x * blockDim.x + threadIdx.x;
    if (e < N_EDGES) atomicAdd(&deg[ei[N_EDGES + e]], 1.0f);   // col = ei[1][e]
}

// Kernel 4: deg -> dinv = rsqrt(deg + 1) in place.
__global__ void k_dinv(float* __restrict__ deg)
{
    int i = blockIdx.x * blockDim.x + threadIdx.x;
    if (i < N_NODES) deg[i] = rsqrtf(deg[i] + 1.0f);
}

// ---------------------------------------------------------------------------
// Kernel 5: xw = x @ W^T via V_WMMA_F32_16X16X4_F32.
// 256 threads = 8 waves; each wave computes a 32x64 tile of xw (two 16-row
// A tiles sharing every B fragment).
//
// ISA 7.12.2 fragment layouts (f32, 16x16x4):
//   A (16x4): lanes 0-15 rows M=0..15 carry {K0,K1}; lanes 16-31 carry {K2,K3}
//   B (4x16): mirrored striping, N across lanes
//   C/D     : VGPR r -> M=r (lanes 0-15) / M=r+8 (lanes 16-31), N = lane%16
//
// LDS is pair-interleaved so each fragment is ONE aligned ds_load_b64:
//   s_wp[p*WP_STRIDE + n*2 + {0,1}] = W[n][2p], W[n][2p+1]
//   s_xp[w][t][p*32 + m*2 + {0,1}]  = x[rowbase_t+m][2p], [2p+1]
// Banking: A pairs -> halves on banks 0-31 / 32-63; B rows padded to 160
// floats (== 32 mod 64) so the two half-waves split banks the same way.
// ---------------------------------------------------------------------------
__global__ void k_gemm_wmma(const float* __restrict__ x,   // [N,64]
                            const float* __restrict__ W,   // [64,64]  W[o][i]
                            float* __restrict__ xw)        // [N,64]
{
    __shared__ float s_wp[32 * WP_STRIDE];   // 20 KB
    __shared__ float s_xp[8][2][1024];       // 64 KB (8 waves x two 16x64 tiles)

    const int tid  = threadIdx.x;
    const int lane = tid & 31;
    const int wid  = tid >> 5;
    const int row0 = blockIdx.x * 256 + wid * 32;   // wave's first x row

    // Stage W pair-interleaved (coalesced float4 global reads).
    for (int i = tid; i < 1024; i += 256) {          // 1024 float4s of W
        int flat = i * 4;
        int o = flat >> 6, k = flat & 63;            // W[o][k..k+3]
        float4 v = *(const float4*)(W + flat);
        int p0 = k >> 1;
        v2f lo; lo.x = v.x; lo.y = v.y;
        v2f hi2; hi2.x = v.z; hi2.y = v.w;
        *(v2f*)&s_wp[(p0 + 0) * WP_STRIDE + o * 2] = lo;
        *(v2f*)&s_wp[(p0 + 1) * WP_STRIDE + o * 2] = hi2;
    }

    // Stage this wave's 32x64 x tile, pair-interleaved. 16 float4 per lane.
    #pragma unroll
    for (int j = 0; j < 16; ++j) {
        int f4  = j * 32 + lane;                     // 0..511
        int off = f4 * 4;
        int r   = off >> 6, cb = off & 63;           // row-in-tile, col base
        float4 v = make_float4(0.f, 0.f, 0.f, 0.f);
        int grow = row0 + r;
        if (grow < N_NODES) v = *(const float4*)(x + (size_t)grow * CH + cb);
        int t  = r >> 4, mm = r & 15, p0 = cb >> 1;
        v2f lo; lo.x = v.x; lo.y = v.y;
        v2f hi2; hi2.x = v.z; hi2.y = v.w;
        *(v2f*)&s_xp[wid][t][(p0 + 0) * 32 + mm * 2] = lo;
        *(v2f*)&s_xp[wid][t][(p0 + 1) * 32 + mm * 2] = hi2;
    }
    __syncthreads();

    const int  mh = lane & 15;
    const bool hi = lane >= 16;

    v8f acc0[4] = {};    // rows row0..row0+15,  n0 = 0,16,32,48
    v8f acc1[4] = {};    // rows row0+16..row0+31

    #pragma unroll
    for (int kk = 0; kk < 16; ++kk) {                // K = 64, 4 per step
        const int p = kk * 2 + (hi ? 1 : 0);         // kpair for this half-wave
        v2f a0 = *(const v2f*)&s_xp[wid][0][p * 32 + mh * 2];
        v2f a1 = *(const v2f*)&s_xp[wid][1][p * 32 + mh * 2];
        #pragma unroll
        for (int g = 0; g < 4; ++g) {
            v2f b = *(const v2f*)&s_wp[p * WP_STRIDE + (g * 16 + mh) * 2];
            acc0[g] = __builtin_amdgcn_wmma_f32_16x16x4_f32(
                false, a0, false, b, (short)0, acc0[g], false, false);
            acc1[g] = __builtin_amdgcn_wmma_f32_16x16x4_f32(
                false, a1, false, b, (short)0, acc1[g], false, false);
        }
    }

    // Store both tiles. Each half-wave writes 64 contiguous bytes per reg row.
    #pragma unroll
    for (int g = 0; g < 4; ++g) {
        const int n = g * 16 + mh;
        #pragma unroll
        for (int r = 0; r < 8; ++r) {
            const int m = r + (hi ? 8 : 0);
            int grow = row0 + m;
            if (grow < N_NODES) xw[(size_t)grow * CH + n] = acc0[g][r];
            grow = row0 + 16 + m;
            if (grow < N_NODES) xw[(size_t)grow * CH + n] = acc1[g][r];
        }
    }
}

// ---------------------------------------------------------------------------
// Kernel 6: edge scatter  out[col] += norm * xw[row].
// 4 threads per edge, 16 channels each (4x float4 gather + 16 f32 atomics).
// ---------------------------------------------------------------------------
__global__ void k_scatter(const int*   __restrict__ ei,
                          const float* __restrict__ xw,
                          const float* __restrict__ dinv,
                          float* __restrict__ out)
{
    int gid = blockIdx.x * blockDim.x + threadIdx.x;
    int e   = gid >> 2;
    if (e >= N_EDGES) return;
    int sub = gid & 3;

    int r = ei[e];                         // source (row)
    int c = ei[N_EDGES + e];               // destination (col)
    float nrm = dinv[r] * dinv[c];

    const float* src = xw  + (size_t)r * CH + sub * 16;
    float*       dst = out + (size_t)c * CH + sub * 16;

    #pragma unroll
    for (int q = 0; q < 4; ++q) {
        float4 v = ((const float4*)src)[q];
        atomicAdd(dst + q * 4 + 0, v.x * nrm);
        atomicAdd(dst + q * 4 + 1, v.y * nrm);
        atomicAdd(dst + q * 4 + 2, v.z * nrm);
        atomicAdd(dst + q * 4 + 3, v.w * nrm);
    }
}

// ---------------------------------------------------------------------------
// Kernel 7: out += xw * dinv^2 (self-loop) + bias. float4 over N*16 quads.
// ---------------------------------------------------------------------------
__global__ void k_finalize(const float* __restrict__ xw,
                           const float* __restrict__ dinv,
                           const float* __restrict__ bias,
                           float* __restrict__ out)
{
    int i4 = blockIdx.x * blockDim.x + threadIdx.x;
    if (i4 >= N_NODES * 16) return;
    int n  = i4 >> 4;
    int cb = (i4 & 15) * 4;
    float di = dinv[n];
    float s  = di * di;

    float4 v = ((const float4*)xw)[i4];
    float4 b = *(const float4*)(bias + cb);
    float4 o = ((float4*)out)[i4];
    o.x += v.x * s + b.x;
    o.y += v.y * s + b.y;
    o.z += v.z * s + b.z;
    o.w += v.w * s + b.w;
    ((float4*)out)[i4] = o;
}

// ---------------------------------------------------------------------------
extern "C" void kernel_launch(void* const* d_in, const int* in_sizes, int n_in,
                              void* d_out, int out_size, void* d_ws, size_t ws_size,
                              hipStream_t stream)
{
    const float* x      = (const float*)d_in[0];
    const int*   ei     = (const int*)  d_in[1];   // [2, N_EDGES]
    const float* mem    = (const float*)d_in[2];
    const float* w_ih   = (const float*)d_in[3];
    // d_in[4] = gru_w_hh: unused (h0 == 0 -> gh = b_hh)
    const float* b_ih   = (const float*)d_in[5];
    const float* b_hh   = (const float*)d_in[6];
    const float* wt_w   = (const float*)d_in[7];
    const float* wt_b   = (const float*)d_in[8];
    const float* bias   = (const float*)d_in[9];
    float*       out    = (float*)d_out;

    // Workspace layout (all 16B aligned):
    //   [0, 16384)                      W   (64x64 f32)
    //   [16384, 16384+400000)           deg -> dinv (N f32)
    //   [416384, 416384+25.6MB)         xw  (N x 64 f32)
    char* ws    = (char*)d_ws;
    float* W    = (float*)(ws);
    float* dinv = (float*)(ws + 16384);
    float* xw   = (float*)(ws + 416384);

    k_gen_weights<<<1, 256, 0, stream>>>(mem, w_ih, b_ih, b_hh, wt_w, wt_b, W);

    {   // zero out + deg
        int n = N_NODES * CH;                       // covers deg range too
        k_zero<<<(n + 255) / 256, 256, 0, stream>>>(out, dinv);
    }

    k_degree<<<(N_EDGES + 255) / 256, 256, 0, stream>>>(ei, dinv);
    k_dinv  <<<(N_NODES + 255) / 256, 256, 0, stream>>>(dinv);

    {   // WMMA GEMM: 256 rows per block (32 per wave)
        int blocks = (N_NODES + 255) / 256;
        k_gemm_wmma<<<blocks, 256, 0, stream>>>(x, W, xw);
    }

    {   // edge scatter: 4 threads / edge
        long long thr = (long long)N_EDGES * 4;
        k_scatter<<<(int)((thr + 255) / 256), 256, 0, stream>>>(ei, xw, dinv, out);
    }

    {   // self-loop + bias
        int q = N_NODES * 16;
        k_finalize<<<(q + 255) / 256, 256, 0, stream>>>(xw, dinv, bias, out);
    }
}